// ObjectDetectionModel_18528488915018
// MI455X (gfx1250) — compile-verified
//
#include <hip/hip_runtime.h>
#include <stdint.h>

// Problem constants (match reference)
#define NB      16
#define NA      25200
#define NCLS    80
#define ND      85          // 5 + NCLS
#define CONF_T  0.25f
#define IOU_T   0.45f
#define TOPK    1024
#define MAXDET  300
#define MAXWH   4096.0f
#define HBINS   4096
#define CAP     2048        // candidate collection capacity per batch
#define NW      32          // 32-bit words per 1024-bit mask

// ---- CDNA5 async global->LDS path (ASYNCcnt), guarded so host pass compiles ----
#if defined(__has_builtin)
#  if __has_builtin(__builtin_amdgcn_global_load_async_to_lds_b128) && \
      __has_builtin(__builtin_amdgcn_s_wait_asynccnt)
#    define HAVE_ASYNC_LDS 1
#  endif
#endif
#ifndef HAVE_ASYNC_LDS
#  define HAVE_ASYNC_LDS 0
#endif

#if HAVE_ASYNC_LDS
typedef int v4i __attribute__((vector_size(16)));
__device__ __forceinline__ void async_copy16_to_lds(const void* g, void* l) {
  // ROCm clang-22 declares this builtin with generic v4i* params
  __builtin_amdgcn_global_load_async_to_lds_b128((v4i*)g, (v4i*)l, 0, 0);
}
#endif

// ---------------- K0: zero histogram + counters ----------------
__global__ void k_zero(int* __restrict__ hist, int* __restrict__ cnt) {
  int i = blockIdx.x * 256 + threadIdx.x;
  if (i < NB * HBINS) hist[i] = 0;
  if (i < NB)         cnt[i]  = 0;
}

// ---------------- K1: decode, conf/cls/boxes, histogram ----------------
__global__ void k_prep(const float* __restrict__ det,
                       float* __restrict__ conf,
                       float* __restrict__ boxes,
                       int*   __restrict__ cls,
                       int*   __restrict__ hist) {
  int idx = blockIdx.x * blockDim.x + threadIdx.x;
  if (idx >= NB * NA) return;
  int b = idx / NA;
  const float* r = det + (size_t)idx * ND;
  float x = r[0], y = r[1], w = r[2], h = r[3], obj = r[4];
  // argmax over products, first-occurrence tie-break (matches jnp.argmax)
  float best = -1.0f; int bc = 0;
  #pragma unroll 4
  for (int c = 0; c < NCLS; ++c) {
    float p = r[5 + c] * obj;
    if (p > best) { best = p; bc = c; }
  }
  float cf = best;
  bool valid = (obj > CONF_T) && (cf > CONF_T);
  cf = valid ? cf : 0.0f;
  float hw = w * 0.5f, hh = h * 0.5f;
  float* bo = boxes + (size_t)idx * 4;
  bo[0] = x - hw; bo[1] = y - hh; bo[2] = x + hw; bo[3] = y + hh;
  conf[idx] = cf;
  cls[idx]  = bc;
  if (cf > 0.0f) {
    int bin = (int)(cf * (float)HBINS);
    bin = bin < 0 ? 0 : (bin > HBINS - 1 ? HBINS - 1 : bin);
    atomicAdd(&hist[b * HBINS + bin], 1);
  }
}

// ---------------- K2: find per-batch threshold bucket ----------------
__global__ void k_thresh(const int* __restrict__ hist, int* __restrict__ tb) {
  int b = blockIdx.x;
  if (threadIdx.x != 0) return;
  int acc = 0, t = 0;
  for (int bin = HBINS - 1; bin >= 0; --bin) {
    acc += hist[b * HBINS + bin];
    if (acc >= TOPK) { t = bin; break; }
  }
  tb[b] = t;
}

// ---------------- K3: collect candidates above threshold ----------------
__global__ void k_collect(const float* __restrict__ conf,
                          const int* __restrict__ tb,
                          int* __restrict__ cnt,
                          float* __restrict__ cand_s,
                          int*   __restrict__ cand_i) {
  int idx = blockIdx.x * blockDim.x + threadIdx.x;
  if (idx >= NB * NA) return;
  int b = idx / NA, i = idx - b * NA;
  float cf = conf[idx];
  if (cf > 0.0f) {
    int bin = (int)(cf * (float)HBINS);
    bin = bin < 0 ? 0 : (bin > HBINS - 1 ? HBINS - 1 : bin);
    if (bin >= tb[b]) {
      int pos = atomicAdd(&cnt[b], 1);
      if (pos < CAP) {
        cand_s[b * CAP + pos] = cf;
        cand_i[b * CAP + pos] = i;
      }
    }
  }
}

// ---------------- K4: bitonic sort (score desc, idx asc) + gather top-1024 ----------------
__global__ void __launch_bounds__(TOPK)
k_sortgather(const float* __restrict__ cand_s,
             const int*   __restrict__ cand_i,
             const int*   __restrict__ cnt,
             const float* __restrict__ boxes,
             const int*   __restrict__ cls,
             float* __restrict__ top_s,
             float* __restrict__ top_c,
             float* __restrict__ top_rb,
             float* __restrict__ top_ob) {
  __shared__ float ss[CAP];
  __shared__ int   si[CAP];
  int b = blockIdx.x, tid = threadIdx.x;
  int n = cnt[b]; if (n > CAP) n = CAP;
  for (int e = tid; e < CAP; e += TOPK) {
    if (e < n) { ss[e] = cand_s[b * CAP + e]; si[e] = cand_i[b * CAP + e]; }
    else       { ss[e] = -1.0f;               si[e] = 0x7fffffff; }
  }
  __syncthreads();
  for (int k = 2; k <= CAP; k <<= 1) {
    for (int j = k >> 1; j > 0; j >>= 1) {
      for (int e = tid; e < CAP; e += TOPK) {
        int p = e ^ j;
        if (p > e) {
          float s0 = ss[e], s1 = ss[p];
          int   i0 = si[e], i1 = si[p];
          bool p_before_e = (s1 > s0) || (s1 == s0 && i1 < i0);
          bool e_before_p = (s0 > s1) || (s0 == s1 && i0 < i1);
          bool sw = ((e & k) == 0) ? p_before_e : e_before_p;
          if (sw) { ss[e] = s1; ss[p] = s0; si[e] = i1; si[p] = i0; }
        }
      }
      __syncthreads();
    }
  }
  if (tid < TOPK) {
    float sc = ss[tid]; int id = si[tid];
    float r0 = 0.f, r1 = 0.f, r2 = 0.f, r3 = 0.f, clf = 0.f, cf = 0.f;
    if (sc > 0.0f) {
      const float* bo = boxes + ((size_t)b * NA + id) * 4;
      r0 = bo[0]; r1 = bo[1]; r2 = bo[2]; r3 = bo[3];
      clf = (float)cls[(size_t)b * NA + id];
      cf = sc;
    }
    float off = clf * MAXWH;
    size_t o = (size_t)b * TOPK + tid;
    top_s[o] = cf; top_c[o] = clf;
    float* rp = top_rb + o * 4;
    rp[0] = r0; rp[1] = r1; rp[2] = r2; rp[3] = r3;
    float* op = top_ob + o * 4;
    op[0] = r0 + off; op[1] = r1 + off; op[2] = r2 + off; op[3] = r3 + off;
  }
}

// ---------------- K5: bitmask NMS (parallel matrix + single-wave reduce) ----------------
__global__ void __launch_bounds__(TOPK)
k_nms(const float* __restrict__ top_s,
      const float* __restrict__ top_c,
      const float* __restrict__ top_rb,
      const float* __restrict__ top_ob,
      float* __restrict__ out) {
  // 320KB LDS per WGP on CDNA5 lets the full 1024x1024-bit suppression
  // matrix (128KB) live on-chip -> phase 2 needs no workgroup barriers.
  __shared__ __attribute__((aligned(16))) float ob[TOPK * 4];   // 16KB
  __shared__ float    area[TOPK];                               // 4KB
  __shared__ float    sc[TOPK];                                 // 4KB
  __shared__ unsigned sup[TOPK * NW];                           // 128KB
  __shared__ unsigned kw[NW];                                   // 128B
  __shared__ unsigned rank[TOPK];                               // 4KB

  int b = blockIdx.x, j = threadIdx.x;
  const float* gob = top_ob + (size_t)b * TOPK * 4;
#if HAVE_ASYNC_LDS
  async_copy16_to_lds(gob + (size_t)j * 4, &ob[j * 4]);
  __builtin_amdgcn_s_wait_asynccnt(0);
#else
  {
    const float4 v = ((const float4*)gob)[j];
    ((float4*)&ob[j * 4])[0] = v;
  }
#endif
  sc[j] = top_s[(size_t)b * TOPK + j];
  __syncthreads();

  float x1 = ob[j * 4], y1 = ob[j * 4 + 1], x2 = ob[j * 4 + 2], y2 = ob[j * 4 + 3];
  float aw = fmaxf(x2 - x1, 0.0f), ah = fmaxf(y2 - y1, 0.0f);
  float ai = aw * ah;
  area[j] = ai;
  __syncthreads();

  // Phase 1: thread j owns suppression row i=j (bits for all columns jj)
  for (int w = 0; w < NW; ++w) {
    unsigned m = 0u;
    int base = w * 32;
    #pragma unroll 8
    for (int t = 0; t < 32; ++t) {
      int jj = base + t;
      float lx = fmaxf(x1, ob[jj * 4]);
      float ly = fmaxf(y1, ob[jj * 4 + 1]);
      float rx = fminf(x2, ob[jj * 4 + 2]);
      float ry = fminf(y2, ob[jj * 4 + 3]);
      float iw = fmaxf(rx - lx, 0.0f), ih = fmaxf(ry - ly, 0.0f);
      float inter = iw * ih;
      float iou = inter / (ai + area[jj] - inter + 1e-7f);
      bool s = (jj > j) && (iou > IOU_T);
      m |= (s ? 1u : 0u) << t;
    }
    sup[j * NW + w] = m;
  }
  __syncthreads();

  // Phase 2: single wave32, lane l owns keep word l (register-resident).
  if (j < NW) {
    unsigned keepw = 0u;
    int base = j * 32;
    for (int t = 0; t < 32; ++t)
      keepw |= (sc[base + t] > 0.0f ? 1u : 0u) << t;
    for (int i = 0; i < TOPK; ++i) {
      unsigned srcw = __shfl(keepw, i >> 5, 32);   // broadcast current keep bit i
      if ((srcw >> (i & 31)) & 1u)
        keepw &= ~sup[i * NW + j];
    }
    kw[j] = keepw;
  }
  __syncthreads();

  unsigned keepbit = (kw[j >> 5] >> (j & 31)) & 1u;

  // inclusive scan (Hillis-Steele) over keep bits
  rank[j] = keepbit;
  __syncthreads();
  for (int off = 1; off < TOPK; off <<= 1) {
    unsigned t = rank[j];
    if (j >= off) t += rank[j - off];
    __syncthreads();
    rank[j] = t;
    __syncthreads();
  }
  unsigned r = rank[j] - keepbit;  // exclusive rank

  float* o = out + (size_t)b * MAXDET * 6;
  for (int t = j; t < MAXDET * 6; t += TOPK) o[t] = 0.0f;
  __threadfence_block();
  __syncthreads();
  if (keepbit && r < (unsigned)MAXDET) {
    size_t q = (size_t)b * TOPK + j;
    const float* rp = top_rb + q * 4;
    float* row = o + (size_t)r * 6;
    row[0] = rp[0]; row[1] = rp[1]; row[2] = rp[2]; row[3] = rp[3];
    row[4] = sc[j]; row[5] = top_c[q];
  }
}

// ---------------- launch ----------------
static inline size_t alup(size_t x) { return (x + 255) & ~(size_t)255; }

extern "C" void kernel_launch(void* const* d_in, const int* in_sizes, int n_in,
                              void* d_out, int out_size, void* d_ws, size_t ws_size,
                              hipStream_t stream) {
  (void)in_sizes; (void)n_in; (void)out_size; (void)ws_size;
  const float* det = (const float*)d_in[0];
  float* out = (float*)d_out;
  char* ws = (char*)d_ws;

  size_t off = 0;
  float* conf   = (float*)(ws + off); off = alup(off + (size_t)NB * NA * 4);
  float* boxes  = (float*)(ws + off); off = alup(off + (size_t)NB * NA * 16);
  int*   cls    = (int*)  (ws + off); off = alup(off + (size_t)NB * NA * 4);
  int*   hist   = (int*)  (ws + off); off = alup(off + (size_t)NB * HBINS * 4);
  int*   tb     = (int*)  (ws + off); off = alup(off + (size_t)NB * 4);
  int*   cnt    = (int*)  (ws + off); off = alup(off + (size_t)NB * 4);
  float* cand_s = (float*)(ws + off); off = alup(off + (size_t)NB * CAP * 4);
  int*   cand_i = (int*)  (ws + off); off = alup(off + (size_t)NB * CAP * 4);
  float* top_s  = (float*)(ws + off); off = alup(off + (size_t)NB * TOPK * 4);
  float* top_c  = (float*)(ws + off); off = alup(off + (size_t)NB * TOPK * 4);
  float* top_rb = (float*)(ws + off); off = alup(off + (size_t)NB * TOPK * 16);
  float* top_ob = (float*)(ws + off); off = alup(off + (size_t)NB * TOPK * 16);

  const int total = NB * NA;
  k_zero<<<(NB * HBINS + 255) / 256, 256, 0, stream>>>(hist, cnt);
  k_prep<<<(total + 255) / 256, 256, 0, stream>>>(det, conf, boxes, cls, hist);
  k_thresh<<<NB, 32, 0, stream>>>(hist, tb);
  k_collect<<<(total + 255) / 256, 256, 0, stream>>>(conf, tb, cnt, cand_s, cand_i);
  k_sortgather<<<NB, TOPK, 0, stream>>>(cand_s, cand_i, cnt, boxes, cls,
                                        top_s, top_c, top_rb, top_ob);
  k_nms<<<NB, TOPK, 0, stream>>>(top_s, top_c, top_rb, top_ob, out);
}